// SRWM_27917287424345
// MI455X (gfx1250) — compile-verified
//
#include <hip/hip_runtime.h>
#include <hip/hip_bf16.h>
#include <math.h>

// ---------------------------------------------------------------------------
// SRWM: fused k/v/q projection GEMM (WMMA f32 16x16x4) + single-wave serial
// fast-weight scan with warm-started power-iteration spectral clamp.
// T=65536, D=256, R=32, M=52.
// ---------------------------------------------------------------------------

typedef __attribute__((ext_vector_type(2))) float v2f;
typedef __attribute__((ext_vector_type(8))) float v8f;

#define T_STEPS 65536
#define DDIM    256
#define RRANK   32
#define MOUT    52

// ---------------------------------------------------------------------------
// Kernel 1: KVQ[t, 0:32]=k, [32:64]=v, [64:96]=q  =  X @ [Wk;Wv;Wq]^T + bias
// One wave computes a 16-row stripe across all 96 output columns, so X is
// read from HBM exactly once (64 MB -> ~3 us at 23.3 TB/s). Weights (96 KB)
// are L2-resident. Uses V_WMMA_F32_16X16X4_F32.
//
// f32 WMMA operand layouts (ISA 7.12.2):
//   A 16x4 : lanes 0-15 -> M=lane, VGPR0/1 = K0/K1 ; lanes 16-31 -> K2/K3
//   B 4x16 : lanes 0-15 -> N=lane, VGPR0/1 = K0/K1 ; lanes 16-31 -> K2/K3
//   D 16x16: VGPR p, lane l -> (M = p + 8*(l>=16), N = l&15)
// Both A and B per-lane fragments are one contiguous 8-byte load.
// ---------------------------------------------------------------------------
__global__ __launch_bounds__(128) void srwm_proj_wmma(
    const float* __restrict__ X,
    const float* __restrict__ Wk, const float* __restrict__ bk,
    const float* __restrict__ Wv, const float* __restrict__ bv,
    const float* __restrict__ Wq, const float* __restrict__ bq,
    float* __restrict__ KVQ) {
  const int lane   = threadIdx.x & 31;
  const int wave   = threadIdx.x >> 5;
  const int tileM  = blockIdx.x * 4 + wave;          // 4096 row tiles
  const int mlo    = lane & 15;
  const int kphase = 2 * (lane >> 4);                // K offset 0 or 2

  const float* Wmat[3] = {Wk, Wv, Wq};
  const float* bvec[3] = {bk, bv, bq};

  v8f acc[6];
#pragma unroll
  for (int nt = 0; nt < 6; ++nt)
    acc[nt] = (v8f){0.f, 0.f, 0.f, 0.f, 0.f, 0.f, 0.f, 0.f};

  const float* aptr = X + (size_t)(tileM * 16 + mlo) * DDIM + kphase;
  const float* bptr[6];
#pragma unroll
  for (int nt = 0; nt < 6; ++nt) {
    const int n = (nt & 1) * 16 + mlo;               // weight row = out col
    bptr[nt] = Wmat[nt >> 1] + (size_t)n * DDIM + kphase;
  }

  for (int k = 0; k < DDIM; k += 4) {
    v2f a = *(const v2f*)(aptr + k);
#pragma unroll
    for (int nt = 0; nt < 6; ++nt) {
      v2f b = *(const v2f*)(bptr[nt] + k);
      acc[nt] = __builtin_amdgcn_wmma_f32_16x16x4_f32(
          /*neg_a=*/false, a, /*neg_b=*/false, b,
          /*c_mod=*/(short)0, acc[nt], /*reuse_a=*/false, /*reuse_b=*/false);
    }
  }

  const int mbase = tileM * 16 + 8 * (lane >> 4);
#pragma unroll
  for (int nt = 0; nt < 6; ++nt) {
    const float bias = bvec[nt >> 1][(nt & 1) * 16 + mlo];
    const int ncol = nt * 16 + mlo;
#pragma unroll
    for (int p = 0; p < 8; ++p)
      KVQ[(size_t)(mbase + p) * 96 + ncol] = acc[nt][p] + bias;
  }
}

// ---------------------------------------------------------------------------
// Log-depth transpose-reduce: lane i holds t_i[0..31]; returns
//   z_lane = sum_i t_i[lane]
// Recursive halving: 5 stages, active array halves each stage (~124 ops vs
// ~320 for a full 5x32 butterfly) -> shorter serial critical path.
// ---------------------------------------------------------------------------
__device__ __forceinline__ float xpose_reduce32(const float t[32], int lane) {
  float a16[16];
  {
    const bool hi = (lane & 16) != 0;
#pragma unroll
    for (int j = 0; j < 16; ++j) {
      float keep = hi ? t[j + 16] : t[j];
      float send = hi ? t[j] : t[j + 16];
      a16[j] = keep + __shfl_xor(send, 16, 32);
    }
  }
  float a8[8];
  {
    const bool hi = (lane & 8) != 0;
#pragma unroll
    for (int j = 0; j < 8; ++j) {
      float keep = hi ? a16[j + 8] : a16[j];
      float send = hi ? a16[j] : a16[j + 8];
      a8[j] = keep + __shfl_xor(send, 8, 32);
    }
  }
  float a4[4];
  {
    const bool hi = (lane & 4) != 0;
#pragma unroll
    for (int j = 0; j < 4; ++j) {
      float keep = hi ? a8[j + 4] : a8[j];
      float send = hi ? a8[j] : a8[j + 4];
      a4[j] = keep + __shfl_xor(send, 4, 32);
    }
  }
  float a2[2];
  {
    const bool hi = (lane & 2) != 0;
#pragma unroll
    for (int j = 0; j < 2; ++j) {
      float keep = hi ? a4[j + 2] : a4[j];
      float send = hi ? a4[j] : a4[j + 2];
      a2[j] = keep + __shfl_xor(send, 2, 32);
    }
  }
  {
    const bool hi = (lane & 1) != 0;
    float keep = hi ? a2[1] : a2[0];
    float send = hi ? a2[0] : a2[1];
    return keep + __shfl_xor(send, 1, 32);
  }
}

// ---------------------------------------------------------------------------
// Kernel 2: serial scan, ONE wave32. Lane i owns row i of W (32 VGPRs).
// Per step:
//   W = (1-d)*W + l*outer(k,v)          (readlane broadcast + 64 FMA)
//   power iter on W^T W (warm start):   y = W u ; z = W^T y (xpose-reduce)
//     sigma ~= ||z||^{1/2}  (u unit)    ; u <- z/||z||  (replicated)
//   spectral clamp: W *= min(1, 1/sigma)
//   fast_out = q @ W  (xpose-reduce)    -> out = 1 + tanh(fast_out @ Wo^T + bo)
// Wo rows `lane` and `lane+32` are register-resident per lane.
// ---------------------------------------------------------------------------
__global__ __launch_bounds__(32) void srwm_scan(
    const float* __restrict__ KVQ, const float* __restrict__ lr,
    const float* __restrict__ decay, const float* __restrict__ Wo,
    const float* __restrict__ bo, float* __restrict__ out) {
  const int lane = threadIdx.x;

  float w[32], u[32];
#pragma unroll
  for (int j = 0; j < 32; ++j) {
    w[j] = 0.f;
    u[j] = (j == 0) ? 1.f : 0.f;   // replicated unit start vector
  }

  // Register-resident output head: lane -> rows (lane) and (lane+32) of Wo.
  float wo0[32], wo1[32];
  const bool has2 = (lane < MOUT - 32);
  const float bo0 = bo[lane];
  const float bo1 = has2 ? bo[lane + 32] : 0.f;
#pragma unroll
  for (int j = 0; j < 32; ++j) {
    wo0[j] = Wo[lane * 32 + j];
    wo1[j] = has2 ? Wo[(lane + 32) * 32 + j] : 0.f;
  }

  for (int t = 0; t < T_STEPS; ++t) {
    const float* p = KVQ + (size_t)t * 96;
    if (t + 16 < T_STEPS) {  // stay ahead of the serial chain
      __builtin_prefetch(KVQ + (size_t)(t + 16) * 96, 0, 0);
      __builtin_prefetch(lr + t + 16, 0, 0);
    }
    const float kk = p[lane];
    const float vv = p[32 + lane];
    const float qq = p[64 + lane];
    const float lt = lr[t];
    const float dt = decay[t];

    // --- Hebbian rank-1 update with decay ---
    const float a = 1.0f - dt;
    const float ck = lt * kk;
#pragma unroll
    for (int j = 0; j < 32; ++j) {
      const float vj = __shfl(vv, j, 32);  // v_readlane broadcast
      w[j] = fmaf(ck, vj, a * w[j]);
    }

    // --- one warm-started power iteration on W^T W ---
    float y = 0.f;
#pragma unroll
    for (int j = 0; j < 32; ++j) y = fmaf(w[j], u[j], y);  // y_i = (W u)_i

    float tv[32];
#pragma unroll
    for (int j = 0; j < 32; ++j) tv[j] = w[j] * y;          // row_i * y_i
    const float z = xpose_reduce32(tv, lane);               // z_l=(W^T y)_l

    float n2 = z * z;                                       // ||z||^2
#pragma unroll
    for (int m = 16; m >= 1; m >>= 1) n2 += __shfl_xor(n2, m, 32);

    const float zn = sqrtf(n2);           // ||z|| ~= sigma_max^2 (u unit)
    const float sigma = sqrtf(zn);
    const bool ok = (n2 > 1e-30f);
    const float rinv = ok ? (1.0f / zn) : 0.f;
#pragma unroll
    for (int j = 0; j < 32; ++j) {        // u <- z/||z||, replicated
      const float zj = __shfl(z, j, 32);
      u[j] = ok ? (zj * rinv) : u[j];
    }

    // --- spectral clamp ---
    const float s = (sigma > 1.0f) ? (1.0f / sigma) : 1.0f;
#pragma unroll
    for (int j = 0; j < 32; ++j) w[j] *= s;

    // --- query against post-update W:  fo_l = (q^T W)_l ---
#pragma unroll
    for (int j = 0; j < 32; ++j) tv[j] = w[j] * qq;
    const float fo = xpose_reduce32(tv, lane);

    // --- fused output head: raw = fo @ Wo^T + bo ; out = 1 + tanh(raw) ---
    float raw0 = bo0, raw1 = bo1;
#pragma unroll
    for (int j = 0; j < 32; ++j) {
      const float foj = __shfl(fo, j, 32);
      raw0 = fmaf(foj, wo0[j], raw0);
      raw1 = fmaf(foj, wo1[j], raw1);
    }
    out[(size_t)t * MOUT + lane] = 1.0f + tanhf(raw0);
    if (has2) out[(size_t)t * MOUT + 32 + lane] = 1.0f + tanhf(raw1);
  }
}

extern "C" void kernel_launch(void* const* d_in, const int* in_sizes, int n_in,
                              void* d_out, int out_size, void* d_ws,
                              size_t ws_size, hipStream_t stream) {
  const float* X  = (const float*)d_in[0];
  const float* lr = (const float*)d_in[1];
  const float* dc = (const float*)d_in[2];
  const float* Wk = (const float*)d_in[3];
  const float* bk = (const float*)d_in[4];
  const float* Wv = (const float*)d_in[5];
  const float* bv = (const float*)d_in[6];
  const float* Wq = (const float*)d_in[7];
  const float* bq = (const float*)d_in[8];
  const float* Wo = (const float*)d_in[9];
  const float* bo = (const float*)d_in[10];
  float* out = (float*)d_out;
  float* KVQ = (float*)d_ws;  // [65536 x 96] f32 = 24 MB scratch

  srwm_proj_wmma<<<T_STEPS / 64, 128, 0, stream>>>(X, Wk, bk, Wv, bv, Wq, bq,
                                                   KVQ);
  srwm_scan<<<1, 32, 0, stream>>>(KVQ, lr, dc, Wo, bo, out);
}